// VectorQuantizerEMA_66623532695732
// MI455X (gfx1250) — compile-verified
//
#include <hip/hip_runtime.h>

// ---------------- problem constants (from reference) ----------------
#define DECAY_F   0.99f
#define OMD_F     0.01f      // 1 - DECAY
#define EPS_F     1e-5f
#define BETA_F    0.25f
#define K_        1024       // num_embeddings
#define D_        128        // embedding_dim
#define N_        32768      // B*H*W = 32*32*32
#define HW_       1024       // H*W
#define BSTRIDE_  131072     // C*H*W floats per batch image (channels-first)

// ---------------- d_out flat offsets (tuple concatenated) ----------------
#define O_ZQ      0                         // [32,128,32,32] = 4194304
#define O_CBL     4194304                   // scalar 0
#define O_COMMIT  4194305                   // scalar
#define O_COMMIT2 4194306                   // scalar
#define O_IDX     4194307                   // [32,32,32] = 32768 (as float)
#define O_NEWE    4227075                   // [1024,128]
#define O_NEWCS   4358147                   // [1024]
#define O_NEWW    4359171                   // [1024,128]

typedef float v2f __attribute__((ext_vector_type(2)));
typedef float v4f __attribute__((ext_vector_type(4)));
typedef float v8f __attribute__((ext_vector_type(8)));

// ============================================================================
// Kernel 1: e2[k] = ||E_k||^2  (one wave per codebook row)  +  zero the
// accumulator regions of d_out (counts / embed-sums / commit) every launch.
// ============================================================================
__global__ void vq_init_kernel(const float* __restrict__ E,
                               float* __restrict__ e2,
                               float* __restrict__ out) {
    const int tid = threadIdx.x;
    const int gid = blockIdx.x * blockDim.x + tid;
    const int nthreads = gridDim.x * blockDim.x;

    // zero accumulators (harness does not re-poison between replays)
    for (int i = gid; i < K_ + K_ * D_; i += nthreads) {
        if (i < K_) out[O_NEWCS + i] = 0.0f;
        else        out[O_NEWW + (i - K_)] = 0.0f;
    }
    if (gid == 0) { out[O_COMMIT] = 0.0f; out[O_CBL] = 0.0f; }

    // one wave32 per E row: lane loads 4 floats (b128), shfl-reduce
    const int wave = gid >> 5;
    const int lane = tid & 31;
    if (wave < K_) {
        const v4f* row = (const v4f*)(E + (size_t)wave * D_);
        v4f v = row[lane];
        float s = v.x * v.x + v.y * v.y + v.z * v.z + v.w * v.w;
        #pragma unroll
        for (int off = 16; off >= 1; off >>= 1) s += __shfl_xor(s, off, 32);
        if (lane == 0) e2[wave] = s;
    }
}

// ============================================================================
// Kernel 2: argmin over codebook via V_WMMA_F32_16X16X4_F32.
//   block = 256 threads (8 waves); each wave owns 16 rows of z; block iterates
//   all 64 column tiles of E (16 codewords each) staged through LDS.
//   dist proxy = e2[k] - 2 * (z . e_k)   (||z||^2 constant per row)
// ============================================================================
#define ETILE_STRIDE 132   // 16B-aligned rows, conflict-free b64 reads

__launch_bounds__(256, 2)
__global__ void vq_argmin_kernel(const float* __restrict__ z_e,
                                 const float* __restrict__ E,
                                 const float* __restrict__ e2,
                                 float* __restrict__ idx_out /* d_out+O_IDX */) {
    __shared__ float ldsE[16 * ETILE_STRIDE];
    __shared__ float ldsE2[16];

    const int tid   = threadIdx.x;
    const int lane  = tid & 31;
    const int wave  = tid >> 5;        // 0..7
    const int lm16  = lane & 15;       // row/col within 16
    const int khalf = lane >> 4;       // 0:k pair {0,1}, 1:k pair {2,3}

    // A fragment rows for this wave (z is channels-first: stride HW_ per c)
    const int rowBase = blockIdx.x * 128 + wave * 16;
    const int n  = rowBase + lm16;
    const int b  = n >> 10;
    const int hw = n & (HW_ - 1);
    const float* zptr = z_e + (size_t)b * BSTRIDE_ + hw;

    // Preload all 32 k-step A fragments (16x4 fp32 per step) into VGPRs.
    // Lane layout per ISA: lanes 0-15 hold K={0,1}, lanes 16-31 hold K={2,3}.
    v2f a[32];
    #pragma unroll
    for (int s = 0; s < 32; ++s) {
        const int k0 = 4 * s + 2 * khalf;
        a[s].x = zptr[(size_t)k0 * HW_];
        a[s].y = zptr[(size_t)(k0 + 1) * HW_];
    }

    float best[8];
    int   bidx[8];
    #pragma unroll
    for (int i = 0; i < 8; ++i) { best[i] = 3.402823e38f; bidx[i] = 0; }

    for (int t = 0; t < 64; ++t) {
        __syncthreads();
        // cooperative stage of E tile [16 codewords x 128] into LDS
        {
            const int r  = tid >> 4;          // 0..15 codeword-in-tile
            const int kc = (tid & 15) * 8;    // 8 floats per thread
            const float* src = E + (size_t)(t * 16 + r) * D_ + kc;
            float* dst = &ldsE[r * ETILE_STRIDE + kc];
            *(v4f*)(dst)     = *(const v4f*)(src);
            *(v4f*)(dst + 4) = *(const v4f*)(src + 4);
            if (tid < 16) ldsE2[tid] = e2[t * 16 + tid];
        }
        __syncthreads();

        if (t + 1 < 64) {  // pull next tile toward the caches (global_prefetch_b8)
            __builtin_prefetch(E + (size_t)((t + 1) * 16 + (tid >> 4)) * D_ +
                               (tid & 15) * 8, 0, 3);
        }

        // C = A(16x128) * B(128x16) via 32 chained 16x16x4 fp32 WMMAs
        v8f acc = {};
        const float* bptr = &ldsE[lm16 * ETILE_STRIDE + 2 * khalf];
        #pragma unroll
        for (int s = 0; s < 32; ++s) {
            v2f bf = *(const v2f*)(bptr + 4 * s);   // ds_load_b64, conflict-free
            acc = __builtin_amdgcn_wmma_f32_16x16x4_f32(
                      false, a[s], false, bf, (short)0, acc, false, false);
        }

        // running argmin: this lane's column is fixed within the tile
        const float e2v = ldsE2[lm16];
        const int   col = t * 16 + lm16;
        #pragma unroll
        for (int i = 0; i < 8; ++i) {
            const float dist = e2v - 2.0f * acc[i];
            if (dist < best[i]) { best[i] = dist; bidx[i] = col; }
        }
    }

    // cross-lane argmin over the 16 lanes sharing each C row
    // (wave32 C layout: VGPR i -> row i for lanes 0-15, row i+8 for lanes 16-31)
    #pragma unroll
    for (int i = 0; i < 8; ++i) {
        float bv = best[i];
        int   bi = bidx[i];
        #pragma unroll
        for (int off = 8; off >= 1; off >>= 1) {
            const float ov = __shfl_xor(bv, off, 16);
            const int   oi = __shfl_xor(bi, off, 16);
            if (ov < bv || (ov == bv && oi < bi)) { bv = ov; bi = oi; }
        }
        if (lm16 == 0) idx_out[rowBase + i + khalf * 8] = (float)bi;
    }
}

// ============================================================================
// Kernel 3: gather z_q = E[idx] (straight-through output == z_q numerically),
// commit-loss accumulation, and segment sums (counts + per-dim sums) via
// atomics into the d_out accumulator regions.
// ============================================================================
__global__ void vq_gather_kernel(const float* __restrict__ z_e,
                                 const float* __restrict__ E,
                                 float* __restrict__ out) {
    __shared__ float red[256];
    const int tid   = threadIdx.x;
    const int nl    = tid & 63;        // latent index within block's 64
    const int cq    = tid >> 6;        // 0..3 channel phase
    const int nbase = blockIdx.x * 64;
    const int b     = nbase >> 10;
    const int hwb   = nbase & (HW_ - 1);

    const int idx = (int)out[O_IDX + nbase + nl];

    float cpart = 0.0f;
    for (int c = cq; c < D_; c += 4) {
        const size_t zo = (size_t)b * BSTRIDE_ + (size_t)c * HW_ + hwb + nl;
        const float z  = z_e[zo];
        const float ev = E[(size_t)idx * D_ + c];
        out[O_ZQ + zo] = ev;                       // coalesced over nl
        const float d = ev - z;
        cpart += d * d;
        atomicAdd(&out[O_NEWW + idx * D_ + c], z); // segment_sum(z)
    }
    if (cq == 0) atomicAdd(&out[O_NEWCS + idx], 1.0f);  // cluster_size

    red[tid] = cpart;
    __syncthreads();
    #pragma unroll
    for (int s = 128; s >= 1; s >>= 1) {
        if (tid < s) red[tid] += red[tid + s];
        __syncthreads();
    }
    if (tid == 0) atomicAdd(&out[O_COMMIT], red[0]);
}

// ============================================================================
// Kernel 4: single 1024-thread block: EMA update, Laplace smoothing,
// new_E = new_w / smoothed, scalar outputs.
// ============================================================================
__launch_bounds__(1024)
__global__ void vq_finalize_kernel(const float* __restrict__ ema_cs,
                                   const float* __restrict__ ema_w,
                                   float* __restrict__ out) {
    __shared__ float sm[K_];
    __shared__ float redsum[K_];
    const int tid = threadIdx.x;

    const float cnt = out[O_NEWCS + tid];                 // raw count
    const float ncs = ema_cs[tid] * DECAY_F + OMD_F * cnt;
    out[O_NEWCS + tid] = ncs;
    redsum[tid] = ncs;
    __syncthreads();
    #pragma unroll
    for (int s = 512; s >= 1; s >>= 1) {
        if (tid < s) redsum[tid] += redsum[tid + s];
        __syncthreads();
    }
    const float nsum = redsum[0];
    sm[tid] = (ncs + EPS_F) / (nsum + (float)K_ * EPS_F) * nsum;
    __syncthreads();

    for (int i = tid; i < K_ * D_; i += 1024) {
        const int k = i >> 7;
        const float sums = out[O_NEWW + i];               // raw segment sum
        const float nw = ema_w[i] * DECAY_F + OMD_F * sums;
        out[O_NEWW + i] = nw;
        out[O_NEWE + i] = nw / sm[k];
    }

    if (tid == 0) {
        const float acc = out[O_COMMIT];                  // raw SSE
        const float commit = BETA_F * acc / (float)((size_t)N_ * D_);
        out[O_CBL]     = 0.0f;
        out[O_COMMIT]  = commit;
        out[O_COMMIT2] = commit;
    }
}

// ============================================================================
extern "C" void kernel_launch(void* const* d_in, const int* in_sizes, int n_in,
                              void* d_out, int out_size, void* d_ws, size_t ws_size,
                              hipStream_t stream) {
    (void)in_sizes; (void)n_in; (void)out_size; (void)ws_size;
    const float* z_e    = (const float*)d_in[0];
    const float* E      = (const float*)d_in[1];
    const float* ema_cs = (const float*)d_in[2];
    const float* ema_w  = (const float*)d_in[3];
    float* out = (float*)d_out;
    float* e2  = (float*)d_ws;   // K_ floats of scratch

    vq_init_kernel  <<<128,       256,  0, stream>>>(E, e2, out);
    vq_argmin_kernel<<<N_ / 128,  256,  0, stream>>>(z_e, E, e2, out + O_IDX);
    vq_gather_kernel<<<N_ / 64,   256,  0, stream>>>(z_e, E, out);
    vq_finalize_kernel<<<1,       1024, 0, stream>>>(ema_cs, ema_w, out);
}